// InteractionLayer_33346126086779
// MI455X (gfx1250) — compile-verified
//
#include <hip/hip_runtime.h>

#define B_   4096
#define F_   32
#define E_   32
#define I_   496
#define IP_  512
#define BN_EPS 1e-5f
#define BE_  (B_ * E_)   // 131072 elements per BN channel

typedef __attribute__((ext_vector_type(2))) float v2f;
typedef __attribute__((ext_vector_type(8))) float v8f;

// ---- compile-time (row, col) pair table for i<j over 32 features ----
struct PairTab { unsigned char r[IP_]; unsigned char c[IP_]; };
static constexpr PairTab make_tab() {
  PairTab t{};
  int k = 0;
  for (int i = 0; i < F_; ++i)
    for (int j = i + 1; j < F_; ++j) { t.r[k] = (unsigned char)i; t.c[k] = (unsigned char)j; ++k; }
  for (; k < IP_; ++k) { t.r[k] = 0; t.c[k] = 0; }
  return t;
}
__constant__ PairTab g_pairs = make_tab();

// ================= Kernel 1: BN statistics per pair-channel =================
// grid = 496 blocks, one channel each; sums over B*E = 131072 products.
__global__ void bn_stats_kernel(const float* __restrict__ xi,
                                const float* __restrict__ xj,
                                double* __restrict__ sums,
                                double* __restrict__ sqs) {
  __shared__ double s1[256];
  __shared__ double s2[256];
  const int i = blockIdx.x;
  const int r = g_pairs.r[i], c = g_pairs.c[i];
  const int t = threadIdx.x;
  double sa = 0.0, sb = 0.0;
  for (int f = t; f < BE_; f += 256) {
    const int bb = f >> 5, e = f & 31;
    const float v = xi[bb * (F_ * E_) + r * E_ + e] * xj[bb * (F_ * E_) + c * E_ + e];
    sa += (double)v;
    sb += (double)v * (double)v;
  }
  s1[t] = sa; s2[t] = sb;
  __syncthreads();
  for (int off = 128; off > 0; off >>= 1) {
    if (t < off) { s1[t] += s1[t + off]; s2[t] += s2[t + off]; }
    __syncthreads();
  }
  if (t == 0) { sums[i] = s1[0]; sqs[i] = s2[0]; }
}

// ============ Kernel 2: fused normalize + WMMA GEMM + softmax + out ============
// One block per batch b. 512 threads = 16 waves.
__global__ void __launch_bounds__(512, 2)
interaction_fused_kernel(const float* __restrict__ xi, const float* __restrict__ xj,
                         const float* __restrict__ W,
                         const float* __restrict__ gamma, const float* __restrict__ beta,
                         const float* __restrict__ attn_w, const float* __restrict__ attn_b,
                         const float* __restrict__ proj_w, const float* __restrict__ proj_b,
                         const double* __restrict__ sums, const double* __restrict__ sqs,
                         float* __restrict__ out) {
  extern __shared__ float smem[];
  float* s_out   = smem;                   // IP_*E_ = 16384 floats (64 KB)
  float* s_xi    = s_out + IP_ * E_;       // 1024
  float* s_xj    = s_xi + F_ * E_;         // 1024
  float* s_logit = s_xj + F_ * E_;         // IP_
  float* s_red   = s_logit + IP_;          // 512

  const int tid = threadIdx.x;
  const int b   = blockIdx.x;
  const float* xib = xi + (size_t)b * (F_ * E_);
  const float* xjb = xj + (size_t)b * (F_ * E_);

  // ---- stage 0: load per-batch xi/xj tiles, init logit accumulators ----
  for (int f = tid; f < F_ * E_; f += 512) { s_xi[f] = xib[f]; s_xj[f] = xjb[f]; }
  s_logit[tid] = 0.0f;
  __syncthreads();

  // ---- stage 1: build normalized, W-scaled interaction rows in LDS ----
  {
    const int i = tid;
    if (i < I_) {
      const int r = g_pairs.r[i], c = g_pairs.c[i];
      const float mean = (float)(sums[i] * (1.0 / (double)BE_));
      const float var  = (float)(sqs[i] * (1.0 / (double)BE_)) - mean * mean;
      const float rs   = rsqrtf(var + BN_EPS);
      const float g    = gamma[i] * rs;
      const float be   = beta[i];
      #pragma unroll
      for (int e = 0; e < E_; ++e) {
        const float v = s_xi[r * E_ + e] * s_xj[c * E_ + e];
        s_out[i * E_ + e] = ((v - mean) * g + be) * W[i * E_ + e];
      }
    } else if (i < IP_) {
      #pragma unroll
      for (int e = 0; e < E_; ++e) s_out[i * E_ + e] = 0.0f;  // zero pad rows 496..511
    }
  }
  __syncthreads();

  // ---- stage 2: h = relu(s_out @ attn_w + attn_b); logits += h . proj_w ----
  // 32 M-tiles x 2 N-tiles of 16x16, K=32 via 8x v_wmma_f32_16x16x4_f32.
  const int lane = tid & 31;
  const int wave = tid >> 5;       // 0..15
  const int half = lane >> 4;      // 0: K=k0..k0+1, 1: K=k0+2..k0+3 (A layout)
  const int l15  = lane & 15;

  for (int t4 = 0; t4 < 4; ++t4) {
    const int T  = wave * 4 + t4;  // 0..63
    const int mt = T >> 1;         // 0..31
    const int nt = T & 1;          // 0..1
    const int nc   = nt * 16 + l15;      // output column a
    const int mrow = mt * 16 + l15;      // A row for this lane

    v8f acc = {};
    #pragma unroll
    for (int k0 = 0; k0 < E_; k0 += 4) {
      const int kk = k0 + 2 * half;
      v2f af, bf;
      af.x = s_out[mrow * E_ + kk];
      af.y = s_out[mrow * E_ + kk + 1];
      bf.x = attn_w[kk * E_ + nc];
      bf.y = attn_w[(kk + 1) * E_ + nc];
      acc = __builtin_amdgcn_wmma_f32_16x16x4_f32(
          /*neg_a=*/false, af, /*neg_b=*/false, bf,
          /*c_mod=*/(short)0, acc, /*reuse_a=*/false, /*reuse_b=*/false);
    }

    const float bias = attn_b[nc];
    const float pw   = proj_w[nc];
    #pragma unroll
    for (int r = 0; r < 8; ++r) {
      float h = acc[r] + bias;
      h = h > 0.0f ? h : 0.0f;
      float v = h * pw;
      // sum over the 16 lanes holding columns of this N-tile
      #pragma unroll
      for (int off = 1; off < 16; off <<= 1) v += __shfl_xor(v, off, 32);
      if (l15 == 0) {
        const int i = mt * 16 + 8 * half + r;   // D layout: M = 8*half + vgpr
        atomicAdd(&s_logit[i], v);              // ds_add_f32
      }
    }
  }
  __syncthreads();

  // ---- stage 3: softmax over the 496 pairs ----
  const float pb = proj_b[0];
  const float lv = (tid < I_) ? (s_logit[tid] + pb) : -3.4e38f;
  s_red[tid] = lv;
  __syncthreads();
  for (int off = 256; off > 0; off >>= 1) {
    if (tid < off) s_red[tid] = fmaxf(s_red[tid], s_red[tid + off]);
    __syncthreads();
  }
  const float mx = s_red[0];
  __syncthreads();
  const float ev = (tid < I_) ? expf(lv - mx) : 0.0f;
  s_red[tid] = ev;
  __syncthreads();
  for (int off = 256; off > 0; off >>= 1) {
    if (tid < off) s_red[tid] += s_red[tid + off];
    __syncthreads();
  }
  const float inv_den = 1.0f / s_red[0];
  __syncthreads();
  if (tid < I_) s_logit[tid] = ev * inv_den;   // scores
  __syncthreads();

  // ---- stage 4: single coalesced output write: scores * out ----
  float* ob = out + (size_t)b * (I_ * E_);
  for (int f = tid; f < I_ * E_; f += 512) {
    ob[f] = s_logit[f >> 5] * s_out[f];
  }
}

extern "C" void kernel_launch(void* const* d_in, const int* in_sizes, int n_in,
                              void* d_out, int out_size, void* d_ws, size_t ws_size,
                              hipStream_t stream) {
  const float* xi     = (const float*)d_in[0];
  const float* xj     = (const float*)d_in[1];
  const float* W      = (const float*)d_in[2];
  const float* gamma  = (const float*)d_in[3];
  const float* beta   = (const float*)d_in[4];
  const float* attn_w = (const float*)d_in[5];
  const float* attn_b = (const float*)d_in[6];
  const float* proj_w = (const float*)d_in[7];
  const float* proj_b = (const float*)d_in[8];
  float* out = (float*)d_out;

  double* sums = (double*)d_ws;
  double* sqs  = sums + I_;

  bn_stats_kernel<<<I_, 256, 0, stream>>>(xi, xj, sums, sqs);

  const size_t shmem = (size_t)(IP_ * E_ + 2 * F_ * E_ + IP_ + 512) * sizeof(float); // 77824 B
  (void)hipFuncSetAttribute((const void*)interaction_fused_kernel,
                            hipFuncAttributeMaxDynamicSharedMemorySize, (int)shmem);
  interaction_fused_kernel<<<B_, 512, shmem, stream>>>(
      xi, xj, W, gamma, beta, attn_w, attn_b, proj_w, proj_b, sums, sqs, out);
}